// DMETM_48541720379872
// MI455X (gfx1250) — compile-verified
//
#include <hip/hip_runtime.h>
#include <math.h>

typedef __attribute__((ext_vector_type(2))) float v2f;
typedef __attribute__((ext_vector_type(8))) float v8f;

#define S_DIM 4
#define K_DIM 50
#define T_DIM 25
#define R_DIM 128
#define V_DIM 10000
#define TH_DIM 800
#define EH_DIM 200
#define B_DIM 128
#define DELTA_F 0.005f

// ---------------- workspace layout (float offsets, all even) ----------------
#define WS_HDR    0LL                      // 16 floats: [0]=ld_p0 [1]=ld_pt [2..5]=diag_p0 [6..9]=diag_pt
#define WS_BETA   16LL                     // T*K*V = 12,500,000
#define WS_XCAT   (WS_BETA + 12500000LL)   // B*(V+K) = 1,286,400
#define WS_H1     (WS_XCAT + 1286400LL)    // B*TH = 102,400
#define WS_H2     (WS_H1 + 102400LL)       // 102,400
#define WS_MUTH   (WS_H2 + 102400LL)       // 6,400
#define WS_LSTH   (WS_MUTH + 6400LL)       // 6,400
#define WS_THETA  (WS_LSTH + 6400LL)       // 6,400
#define WS_ETAINP (WS_THETA + 6400LL)      // T*EH = 5,000
#define WS_HSEQ   (WS_ETAINP + 5000LL)     // 5,000
#define WS_ETAS   (WS_HSEQ + 5000LL)       // T*K = 1,250

static __device__ __forceinline__ v8f wmma4(v2f a, v2f b, v8f c) {
    return __builtin_amdgcn_wmma_f32_16x16x4_f32(false, a, false, b, (short)0, c, false, false);
}

// ============================================================================
// Batched GEMM-NT via f32 WMMA: C[b][m][n] = sum_d A[b][m][d]*B[n][d] (+bias,
// optional tanh).  One wave computes a 32x64 tile (2 M-frags x 4 N-frags):
// per 4-wide K chunk -> 6 aligned b64 loads, 8 v_wmma_f32_16x16x4_f32.
// Edge rows/cols are CLAMPED for loads (always-valid addresses, no exec-mask
// churn); correctness preserved because clamped lanes only feed output
// elements that the store-side guard drops.  K remainder handled once.
// Requires lda/ldb and A/B base element offsets even (true for all uses).
// ============================================================================
__global__ __launch_bounds__(256)
void gemm_nt_wmma(const float* __restrict__ A, long long lda, long long aBatch,
                  const float* __restrict__ Bm, long long ldb,
                  const float* __restrict__ bias,
                  float* __restrict__ C, long long ldc, long long cBatch,
                  int M, int N, int Kd, int act)
{
    const int lane = threadIdx.x & 31;
    const int wave = threadIdx.x >> 5;
    const int tilesM2 = (M + 31) >> 5;      // 32-row groups
    const int tilesN4 = (N + 63) >> 6;      // 64-col groups
    const int tile = blockIdx.x * 8 + wave;
    if (tile >= tilesM2 * tilesN4) return;  // wave-uniform exit: EXEC all-1 for WMMA
    const int tm2 = tile / tilesN4;
    const int tn4 = tile % tilesN4;

    const float* Ab = A + (long long)blockIdx.y * aBatch;
    float*       Cb = C + (long long)blockIdx.y * cBatch;

    const int l15 = lane & 15;
    const int kk  = (lane >> 4) * 2;        // K sub-offset within the 4-chunk

    // A fragments: rows m0 (tile 0) and m0+16 (tile 1), clamped for loads
    const int m0 = tm2 * 32 + l15;
    const int m1 = m0 + 16;
    const long long arow0 = (long long)(m0 < M ? m0 : M - 1) * lda + kk;
    const long long arow1 = (long long)(m1 < M ? m1 : M - 1) * lda + kk;

    // B fragments: 4 consecutive 16-col groups, clamped for loads
    long long brow[4];
#pragma unroll
    for (int j = 0; j < 4; ++j) {
        const int nc = tn4 * 64 + j * 16 + l15;
        brow[j] = (long long)(nc < N ? nc : N - 1) * ldb + kk;
    }

    v8f acc[2][4];
#pragma unroll
    for (int i = 0; i < 2; ++i)
#pragma unroll
        for (int j = 0; j < 4; ++j)
            acc[i][j] = (v8f){0.f, 0.f, 0.f, 0.f, 0.f, 0.f, 0.f, 0.f};

    const int KdMain = Kd & ~3;
    for (int r0 = 0; r0 < KdMain; r0 += 4) {
        if ((r0 & 31) == 0 && r0 + 260 < KdMain) {      // uniform branch
#pragma unroll
            for (int j = 0; j < 4; ++j)
                __builtin_prefetch(Bm + brow[j] + r0 + 256, 0, 0);
        }
        const v2f a0 = *(const v2f*)(Ab + arow0 + r0);
        const v2f a1 = *(const v2f*)(Ab + arow1 + r0);
        v2f b[4];
#pragma unroll
        for (int j = 0; j < 4; ++j) b[j] = *(const v2f*)(Bm + brow[j] + r0);
#pragma unroll
        for (int j = 0; j < 4; ++j) {
            acc[0][j] = wmma4(a0, b[j], acc[0][j]);
            acc[1][j] = wmma4(a1, b[j], acc[1][j]);
        }
    }
    if (KdMain < Kd) {                      // K remainder (1..3), runs once
        const int ra = KdMain + kk;
        v2f a0, a1, b[4];
        a0.x = (ra     < Kd) ? Ab[arow0 + KdMain]     : 0.f;
        a0.y = (ra + 1 < Kd) ? Ab[arow0 + KdMain + 1] : 0.f;
        a1.x = (ra     < Kd) ? Ab[arow1 + KdMain]     : 0.f;
        a1.y = (ra + 1 < Kd) ? Ab[arow1 + KdMain + 1] : 0.f;
#pragma unroll
        for (int j = 0; j < 4; ++j) {
            b[j].x = (ra     < Kd) ? Bm[brow[j] + KdMain]     : 0.f;
            b[j].y = (ra + 1 < Kd) ? Bm[brow[j] + KdMain + 1] : 0.f;
        }
#pragma unroll
        for (int j = 0; j < 4; ++j) {
            acc[0][j] = wmma4(a0, b[j], acc[0][j]);
            acc[1][j] = wmma4(a1, b[j], acc[1][j]);
        }
    }

    // Store: C/D layout -> vgpr i holds row (i + 8*(lane>=16)), col = lane&15
    const int rsub = (lane >> 4) << 3;
#pragma unroll
    for (int j = 0; j < 4; ++j) {
        const int colv = tn4 * 64 + j * 16 + l15;
        if (colv < N) {
            const float bs = bias ? bias[colv] : 0.f;
#pragma unroll
            for (int i2 = 0; i2 < 2; ++i2) {
                const int rbase = tm2 * 32 + i2 * 16 + rsub;
#pragma unroll
                for (int i = 0; i < 8; ++i) {
                    const int row = rbase + i;
                    if (row < M) {
                        float v = acc[i2][j][i] + bs;
                        if (act == 1) v = tanhf(v);
                        Cb[(long long)row * ldc + colv] = v;
                    }
                }
            }
        }
    }
}

// ============================================================================
// Setup: zero the 5 output accumulators; slogdet + diagonals of the two 4x4
// prior covariances (Sp0 = exp(slc), Spt = DELTA*slc) via LU w/ pivoting.
// ============================================================================
__device__ double logabsdet4(double a[4][4]) {
    double ld = 0.0;
    for (int p = 0; p < 4; ++p) {
        int piv = p; double mx = fabs(a[p][p]);
        for (int r = p + 1; r < 4; ++r)
            if (fabs(a[r][p]) > mx) { mx = fabs(a[r][p]); piv = r; }
        if (piv != p)
            for (int c = 0; c < 4; ++c) { double t = a[p][c]; a[p][c] = a[piv][c]; a[piv][c] = t; }
        ld += log(fabs(a[p][p]));
        for (int r = p + 1; r < 4; ++r) {
            double f = a[r][p] / a[p][p];
            for (int c = p; c < 4; ++c) a[r][c] -= f * a[p][c];
        }
    }
    return ld;
}

__global__ void setup_kernel(const float* __restrict__ slc,
                             float* __restrict__ hdr, float* __restrict__ outp)
{
    if (threadIdx.x == 0 && blockIdx.x == 0) {
        for (int i = 0; i < 5; ++i) outp[i] = 0.f;
        double m0[4][4], mt[4][4];
        for (int i = 0; i < 4; ++i)
            for (int j = 0; j < 4; ++j) {
                double v = (double)slc[i * 4 + j];
                m0[i][j] = exp(v);
                mt[i][j] = (double)DELTA_F * v;
            }
        for (int i = 0; i < 4; ++i) { hdr[2 + i] = (float)m0[i][i]; hdr[6 + i] = (float)mt[i][i]; }
        hdr[0] = (float)logabsdet4(m0);
        hdr[1] = (float)logabsdet4(mt);
    }
}

// ============================================================================
// kl_alpha: fully parallel reduction over (k,t,r); matches reference exactly.
// ============================================================================
__global__ __launch_bounds__(256)
void kl_alpha_kernel(const float* __restrict__ mu, const float* __restrict__ ls,
                     const float* __restrict__ hdr, float* __restrict__ outp)
{
    __shared__ float red[256];
    const int total = K_DIM * T_DIM * R_DIM;
    const float ld_p0 = hdr[0], ld_pt = hdr[1];
    float acc = 0.f;
    for (int idx = blockIdx.x * 256 + threadIdx.x; idx < total; idx += gridDim.x * 256) {
        const int r = idx % R_DIM;
        const int t = (idx / R_DIM) % T_DIM;
        const int k = idx / (R_DIM * T_DIM);
        float logdet_q = 0.f, tr = 0.f, quad = 0.f;
#pragma unroll
        for (int s = 0; s < S_DIM; ++s) {
            const long long off = (((long long)s * K_DIM + k) * T_DIM + t) * R_DIM + r;
            const float lsv = ls[off];
            const float inv = expf(-lsv);
            logdet_q += lsv;
            tr += ((t == 0) ? hdr[2 + s] : hdr[6 + s]) * inv;
            float m = mu[off];
            if (t > 0) m -= mu[off - R_DIM];          // alpha_{t-1} prior mean
            quad += m * m * inv;
        }
        acc += logdet_q - ((t == 0) ? ld_p0 : ld_pt) - (float)S_DIM + tr + quad;
    }
    red[threadIdx.x] = acc; __syncthreads();
    for (int s = 128; s > 0; s >>= 1) {
        if (threadIdx.x < s) red[threadIdx.x] += red[threadIdx.x + s];
        __syncthreads();
    }
    if (threadIdx.x == 0) atomicAdd(&outp[2], red[0]);
}

// ============================================================================
// Sequential part: LSTM over T steps, then eta Markov posterior scan + kl_eta.
// Single 256-thread block (tiny recurrent state: EH=200, K=50).
// ============================================================================
__global__ __launch_bounds__(256)
void recurrent_kernel(const float* __restrict__ eta_inp,
                      const float* __restrict__ Wih, const float* __restrict__ Whh,
                      const float* __restrict__ bih, const float* __restrict__ bhh,
                      const float* __restrict__ muW, const float* __restrict__ mub,
                      const float* __restrict__ lsW, const float* __restrict__ lsb,
                      float* __restrict__ h_seq, float* __restrict__ etas,
                      float* __restrict__ outp)
{
    __shared__ float h[EH_DIM], c[EH_DIM], g[4 * EH_DIM], inp[EH_DIM + K_DIM], red[256];
    const int tid = threadIdx.x;

    for (int u = tid; u < EH_DIM; u += 256) { h[u] = 0.f; c[u] = 0.f; }
    __syncthreads();

    // ---- LSTM (torch gate order i,f,g,o) ----
    for (int t = 0; t < T_DIM; ++t) {
        const float* xt = eta_inp + (long long)t * EH_DIM;
        for (int j = tid; j < 4 * EH_DIM; j += 256) {
            const float* wi = Wih + (long long)j * EH_DIM;
            const float* wh = Whh + (long long)j * EH_DIM;
            float s = bih[j] + bhh[j];
            for (int u = 0; u < EH_DIM; ++u) s += wi[u] * xt[u] + wh[u] * h[u];
            g[j] = s;
        }
        __syncthreads();
        for (int u = tid; u < EH_DIM; u += 256) {
            const float ig = 1.f / (1.f + expf(-g[u]));
            const float fg = 1.f / (1.f + expf(-g[EH_DIM + u]));
            const float gg = tanhf(g[2 * EH_DIM + u]);
            const float og = 1.f / (1.f + expf(-g[3 * EH_DIM + u]));
            const float cn = fg * c[u] + ig * gg;
            c[u] = cn;
            const float hn = og * tanhf(cn);
            h[u] = hn;
            h_seq[(long long)t * EH_DIM + u] = hn;
        }
        __syncthreads();
    }

    // ---- eta scan: mu_t = W[h_t; eta_{t-1}] + b; eval mode eta_t = mu_t ----
    float kl_acc = 0.f;
    for (int j = tid; j < K_DIM; j += 256) inp[EH_DIM + j] = 0.f;
    __syncthreads();
    for (int t = 0; t < T_DIM; ++t) {
        for (int u = tid; u < EH_DIM; u += 256) inp[u] = h_seq[(long long)t * EH_DIM + u];
        __syncthreads();
        float mu_r = 0.f;
        const float pls = (t == 0) ? 0.f : logf(DELTA_F);
        const float denom = expf(pls) + 1e-6f;
        if (tid < K_DIM) {
            const float* wm = muW + (long long)tid * (EH_DIM + K_DIM);
            const float* wl = lsW + (long long)tid * (EH_DIM + K_DIM);
            float sm = mub[tid], sl = lsb[tid];
            for (int u = 0; u < EH_DIM + K_DIM; ++u) {
                const float x = inp[u];
                sm += wm[u] * x; sl += wl[u] * x;
            }
            mu_r = sm;
            const float ep = inp[EH_DIM + tid];
            kl_acc += 0.5f * ((expf(sl) + (sm - ep) * (sm - ep)) / denom - 1.f + pls - sl);
        }
        __syncthreads();
        if (tid < K_DIM) { inp[EH_DIM + tid] = mu_r; etas[(long long)t * K_DIM + tid] = mu_r; }
        __syncthreads();
    }
    red[tid] = kl_acc; __syncthreads();
    for (int s = 128; s > 0; s >>= 1) {
        if (tid < s) red[tid] += red[tid + s];
        __syncthreads();
    }
    if (tid == 0) atomicAdd(&outp[3], red[0]);
}

// ============================================================================
// Build concatenated theta-net input x = [normalized_bows | eta_{times[b]}]
// ============================================================================
__global__ void xcat_kernel(const float* __restrict__ nbows, const float* __restrict__ etas,
                            const int* __restrict__ times, float* __restrict__ x)
{
    const long long idx = (long long)blockIdx.x * blockDim.x + threadIdx.x;
    const long long total = (long long)B_DIM * (V_DIM + K_DIM);
    if (idx >= total) return;
    const int b = (int)(idx / (V_DIM + K_DIM));
    const int d = (int)(idx % (V_DIM + K_DIM));
    x[idx] = (d < V_DIM) ? nbows[(long long)b * V_DIM + d]
                         : etas[(long long)times[b] * K_DIM + (d - V_DIM)];
}

// theta = softmax(mu_th) per doc row (K=50)
__global__ void softmax_theta(const float* __restrict__ mu_th, float* __restrict__ theta)
{
    __shared__ float buf[64];
    const int b = blockIdx.x, tid = threadIdx.x;
    const float v = (tid < K_DIM) ? mu_th[(long long)b * K_DIM + tid] : -__builtin_inff();
    buf[tid] = v; __syncthreads();
    for (int s = 32; s > 0; s >>= 1) { if (tid < s) buf[tid] = fmaxf(buf[tid], buf[tid + s]); __syncthreads(); }
    const float mx = buf[0]; __syncthreads();
    const float e = (tid < K_DIM) ? expf(v - mx) : 0.f;
    buf[tid] = e; __syncthreads();
    for (int s = 32; s > 0; s >>= 1) { if (tid < s) buf[tid] += buf[tid + s]; __syncthreads(); }
    if (tid < K_DIM) theta[(long long)b * K_DIM + tid] = e / buf[0];
}

// kl_theta (scaled by num_docs/B)
__global__ __launch_bounds__(256)
void kl_theta_kernel(const float* __restrict__ mu_th, const float* __restrict__ ls_th,
                     const float* __restrict__ etas, const int* __restrict__ times,
                     const int* __restrict__ ndocs, float* __restrict__ outp)
{
    __shared__ float red[256];
    float acc = 0.f;
    for (int idx = blockIdx.x * 256 + threadIdx.x; idx < B_DIM * K_DIM; idx += gridDim.x * 256) {
        const int b = idx / K_DIM, k = idx % K_DIM;
        const float m = mu_th[idx], l = ls_th[idx];
        const float e = etas[(long long)times[b] * K_DIM + k];
        acc += 0.5f * ((expf(l) + (m - e) * (m - e)) / (1.f + 1e-6f) - 1.f - l);
    }
    red[threadIdx.x] = acc; __syncthreads();
    for (int s = 128; s > 0; s >>= 1) {
        if (threadIdx.x < s) red[threadIdx.x] += red[threadIdx.x + s];
        __syncthreads();
    }
    if (threadIdx.x == 0) {
        const float coeff = (float)(*ndocs) / (float)B_DIM;
        atomicAdd(&outp[4], red[0] * coeff);
    }
}

// beta rows: in-place softmax over V per (t,k) row
__global__ __launch_bounds__(256)
void softmax_rows(float* __restrict__ logits)
{
    __shared__ float red[256];
    const long long base = (long long)blockIdx.x * V_DIM;
    const int tid = threadIdx.x;
    float mx = -__builtin_inff();
    for (int v = tid; v < V_DIM; v += 256) mx = fmaxf(mx, logits[base + v]);
    red[tid] = mx; __syncthreads();
    for (int s = 128; s > 0; s >>= 1) { if (tid < s) red[tid] = fmaxf(red[tid], red[tid + s]); __syncthreads(); }
    mx = red[0]; __syncthreads();
    float sum = 0.f;
    for (int v = tid; v < V_DIM; v += 256) sum += expf(logits[base + v] - mx);
    red[tid] = sum; __syncthreads();
    for (int s = 128; s > 0; s >>= 1) { if (tid < s) red[tid] += red[tid + s]; __syncthreads(); }
    const float inv = 1.f / red[0];
    for (int v = tid; v < V_DIM; v += 256) logits[base + v] = expf(logits[base + v] - mx) * inv;
}

// nll: one block per doc; mix over K topics, log, dot with bows
__global__ __launch_bounds__(256)
void nll_kernel(const float* __restrict__ theta, const float* __restrict__ beta,
                const float* __restrict__ bows, const int* __restrict__ times,
                const int* __restrict__ ndocs, float* __restrict__ outp)
{
    __shared__ float th[K_DIM];
    __shared__ float red[256];
    const int b = blockIdx.x, tid = threadIdx.x;
    const int t = times[b];
    if (tid < K_DIM) th[tid] = theta[(long long)b * K_DIM + tid];
    __syncthreads();
    const float* bt = beta + (long long)t * K_DIM * V_DIM;
    float acc = 0.f;
    for (int v = tid; v < V_DIM; v += 256) {
        float mix = 0.f;
#pragma unroll 10
        for (int k = 0; k < K_DIM; ++k) mix += th[k] * bt[(long long)k * V_DIM + v];
        acc += bows[(long long)b * V_DIM + v] * logf(mix + 1e-6f);
    }
    red[tid] = acc; __syncthreads();
    for (int s = 128; s > 0; s >>= 1) { if (tid < s) red[tid] += red[tid + s]; __syncthreads(); }
    if (tid == 0) {
        const float coeff = (float)(*ndocs) / (float)B_DIM;
        atomicAdd(&outp[1], -red[0] * coeff);
    }
}

__global__ void final_kernel(float* __restrict__ outp)
{
    if (threadIdx.x == 0 && blockIdx.x == 0)
        outp[0] = outp[1] + outp[2] + outp[3] + outp[4];
}

// ============================================================================
static inline void launch_gemm(const float* A, long long lda, long long aB,
                               const float* Bm, long long ldb, const float* bias,
                               float* C, long long ldc, long long cB,
                               int M, int N, int Kd, int act, int batch, hipStream_t s)
{
    const int tiles = ((M + 31) / 32) * ((N + 63) / 64);
    dim3 grid((tiles + 7) / 8, batch);
    gemm_nt_wmma<<<grid, 256, 0, s>>>(A, lda, aB, Bm, ldb, bias, C, ldc, cB, M, N, Kd, act);
}

extern "C" void kernel_launch(void* const* d_in, const int* in_sizes, int n_in,
                              void* d_out, int out_size, void* d_ws, size_t ws_size,
                              hipStream_t stream)
{
    (void)in_sizes; (void)n_in; (void)out_size; (void)ws_size;
    const float* mu_q     = (const float*)d_in[0];
    const float* ls_q     = (const float*)d_in[1];
    const float* slc      = (const float*)d_in[2];
    const float* rho_W    = (const float*)d_in[3];
    const float* th_W1    = (const float*)d_in[4];
    const float* th_b1    = (const float*)d_in[5];
    const float* th_W2    = (const float*)d_in[6];
    const float* th_b2    = (const float*)d_in[7];
    const float* muth_W   = (const float*)d_in[8];
    const float* muth_b   = (const float*)d_in[9];
    const float* lsth_W   = (const float*)d_in[10];
    const float* lsth_b   = (const float*)d_in[11];
    const float* emap_W   = (const float*)d_in[12];
    const float* emap_b   = (const float*)d_in[13];
    const float* Wih      = (const float*)d_in[14];
    const float* Whh      = (const float*)d_in[15];
    const float* bih      = (const float*)d_in[16];
    const float* bhh      = (const float*)d_in[17];
    const float* mueta_W  = (const float*)d_in[18];
    const float* mueta_b  = (const float*)d_in[19];
    const float* lseta_W  = (const float*)d_in[20];
    const float* lseta_b  = (const float*)d_in[21];
    const float* bows     = (const float*)d_in[22];
    const float* nbows    = (const float*)d_in[23];
    const float* rnn_inp  = (const float*)d_in[24];
    const int*   times    = (const int*)d_in[25];
    const int*   ndocs    = (const int*)d_in[26];

    float* out = (float*)d_out;
    float* ws  = (float*)d_ws;
    float* hdr    = ws + WS_HDR;
    float* beta   = ws + WS_BETA;
    float* xcat   = ws + WS_XCAT;
    float* h1     = ws + WS_H1;
    float* h2     = ws + WS_H2;
    float* mu_th  = ws + WS_MUTH;
    float* ls_th  = ws + WS_LSTH;
    float* theta  = ws + WS_THETA;
    float* etainp = ws + WS_ETAINP;
    float* hseq   = ws + WS_HSEQ;
    float* etas   = ws + WS_ETAS;

    // 1) zero outputs + 4x4 slogdets
    setup_kernel<<<1, 32, 0, stream>>>(slc, hdr, out);

    // 2) kl_alpha reduction over (k,t,r)
    kl_alpha_kernel<<<625, 256, 0, stream>>>(mu_q, ls_q, hdr, out);

    // 3) eta_inp = rnn_inp @ eta_map_W^T + b   (25 x 200, Kd=10000)
    launch_gemm(rnn_inp, V_DIM, 0, emap_W, V_DIM, emap_b,
                etainp, EH_DIM, 0, T_DIM, EH_DIM, V_DIM, 0, 1, stream);

    // 4) LSTM + eta scan (+ kl_eta)
    recurrent_kernel<<<1, 256, 0, stream>>>(etainp, Wih, Whh, bih, bhh,
                                            mueta_W, mueta_b, lseta_W, lseta_b,
                                            hseq, etas, out);

    // 5) x = [normalized_bows | eta_td]
    {
        const long long total = (long long)B_DIM * (V_DIM + K_DIM);
        xcat_kernel<<<(unsigned)((total + 255) / 256), 256, 0, stream>>>(nbows, etas, times, xcat);
    }

    // 6) theta MLP: h1 = tanh(x W1^T + b1); h2 = tanh(h1 W2^T + b2); heads
    launch_gemm(xcat, V_DIM + K_DIM, 0, th_W1, V_DIM + K_DIM, th_b1,
                h1, TH_DIM, 0, B_DIM, TH_DIM, V_DIM + K_DIM, 1, 1, stream);
    launch_gemm(h1, TH_DIM, 0, th_W2, TH_DIM, th_b2,
                h2, TH_DIM, 0, B_DIM, TH_DIM, TH_DIM, 1, 1, stream);
    launch_gemm(h2, TH_DIM, 0, muth_W, TH_DIM, muth_b,
                mu_th, K_DIM, 0, B_DIM, K_DIM, TH_DIM, 0, 1, stream);
    launch_gemm(h2, TH_DIM, 0, lsth_W, TH_DIM, lsth_b,
                ls_th, K_DIM, 0, B_DIM, K_DIM, TH_DIM, 0, 1, stream);

    softmax_theta<<<B_DIM, 64, 0, stream>>>(mu_th, theta);
    kl_theta_kernel<<<25, 256, 0, stream>>>(mu_th, ls_th, etas, times, ndocs, out);

    // 7) beta logits for s=0, batched over t: C[t][k][v] = alpha0[t,k,:]·rho[v,:]
    //    A[k][d] for batch t = mu_q[(k*T + t)*R + d]  -> base=mu_q, lda=T*R, batch stride=R
    launch_gemm(mu_q, (long long)T_DIM * R_DIM, R_DIM, rho_W, R_DIM, nullptr,
                beta, V_DIM, (long long)K_DIM * V_DIM,
                K_DIM, V_DIM, R_DIM, 0, T_DIM, stream);

    // 8) softmax over V per (t,k) row -> beta probabilities (in place)
    softmax_rows<<<T_DIM * K_DIM, 256, 0, stream>>>(beta);

    // 9) nll per doc
    nll_kernel<<<B_DIM, 256, 0, stream>>>(theta, beta, bows, times, ndocs, out);

    // 10) nelbo = nll + kl_alpha + kl_eta + kl_theta
    final_kernel<<<1, 32, 0, stream>>>(out);
}